// Decoder_87651692576924
// MI455X (gfx1250) — compile-verified
//
#include <hip/hip_runtime.h>
#include <math.h>

#define NGRAPHS 50000
#define INFEAT 32
#define FFN 128
#define HIDDEN 16
#define CW 256                 // conv channel width
#define NEDGES 800000

typedef float v2f __attribute__((ext_vector_type(2)));
typedef float v4f __attribute__((ext_vector_type(4)));
typedef float v8f __attribute__((ext_vector_type(8)));

// D = A(16x4 f32) * B(4x16 f32) + C(16x16 f32)  -> v_wmma_f32_16x16x4_f32
__device__ __forceinline__ v8f wmma4(v2f a, v2f b, v8f c) {
    return __builtin_amdgcn_wmma_f32_16x16x4_f32(false, a, false, b, (short)0, c,
                                                 false, false);
}

__device__ __forceinline__ float elu1(float t) {
    return t > 0.f ? t : (__expf(t) - 1.f);
}

// ---------------------------------------------------------------------------
// Weight transpose: WT[n*K + k] = W[k*N + n]  (so B-fragments load as b64)
// ---------------------------------------------------------------------------
__global__ void gnn_transpose_kernel(const float* __restrict__ W,
                                     float* __restrict__ WT, int K, int N) {
    int i = blockIdx.x * blockDim.x + threadIdx.x;
    if (i < K * N) {
        int k = i / N, n = i - k * N;
        WT[n * K + k] = W[i];
    }
}

// ---------------------------------------------------------------------------
// Fused MLP: h = elu((x @ fc2_w + fc2_b) @ fc1_w + fc1_b) -> flat [50000,256]
// One wave per 16 rows, weights pre-transposed (b64 B loads).
// ---------------------------------------------------------------------------
__global__ __launch_bounds__(128) void gnn_mlp_kernel(
    const float* __restrict__ x, const float* __restrict__ w2T,  // [128][32]
    const float* __restrict__ b2, const float* __restrict__ w1T, // [16][128]
    const float* __restrict__ b1, float* __restrict__ h0) {
    __shared__ float lds[4][16 * FFN];   // 32 KB / block
    const int wid  = threadIdx.x >> 5;
    const int lane = threadIdx.x & 31;
    const int half = lane >> 4;
    const int lr   = lane & 15;
    const int tile = blockIdx.x * 4 + wid;   // 0..49999
    const int r0   = tile * 16;

    float* tbuf = lds[wid];

    // fc2: [16x32] @ [32x128] + b2 -> LDS (16x128 row-major)
    #pragma unroll
    for (int nt = 0; nt < 8; ++nt) {
        const int cn = nt * 16;
        v8f acc;
        const float bb = b2[cn + lr];
        #pragma unroll
        for (int v = 0; v < 8; ++v) acc[v] = bb;
        #pragma unroll
        for (int k = 0; k < INFEAT; k += 4) {
            const int kk = k + 2 * half;
            v2f a = *(const v2f*)(x + (size_t)(r0 + lr) * INFEAT + kk);
            v2f b = *(const v2f*)(w2T + (cn + lr) * INFEAT + kk);
            acc = wmma4(a, b, acc);
        }
        #pragma unroll
        for (int v = 0; v < 8; ++v)
            tbuf[(v + 8 * half) * FFN + cn + lr] = acc[v];
    }
    // wave-private LDS region; DS ops are in-order within a wave

    // fc1: [16x128] @ [128x16] + b1, ELU
    v8f acc;
    const float bb = b1[lr];
    #pragma unroll
    for (int v = 0; v < 8; ++v) acc[v] = bb;
    for (int k = 0; k < FFN; k += 4) {
        const int kk = k + 2 * half;
        v2f a = *(const v2f*)(tbuf + lr * FFN + kk);
        v2f b = *(const v2f*)(w1T + lr * FFN + kk);
        acc = wmma4(a, b, acc);
    }
    #pragma unroll
    for (int v = 0; v < 8; ++v)
        h0[(size_t)(r0 + v + 8 * half) * HIDDEN + lr] = elu1(acc[v]);
}

// ---------------------------------------------------------------------------
// GEMM: Out[50000,256] = A[50000,256] @ W (W given transposed [N][K]).
// Block = 4 waves = one 16-row M tile over full N; wave does 16x64.
// ---------------------------------------------------------------------------
__global__ __launch_bounds__(128) void gnn_gemm256_kernel(
    const float* __restrict__ A, const float* __restrict__ WT,
    float* __restrict__ Out) {
    const int wid  = threadIdx.x >> 5;
    const int lane = threadIdx.x & 31;
    const int half = lane >> 4;
    const int lr   = lane & 15;
    const int rm   = blockIdx.x * 16;
    const int cn0  = wid * 64;

    v8f acc[4];
    #pragma unroll
    for (int t = 0; t < 4; ++t)
        #pragma unroll
        for (int v = 0; v < 8; ++v) acc[t][v] = 0.f;

    const float* arow = A + (size_t)(rm + lr) * CW;
    for (int k = 0; k < CW; k += 4) {
        const int kk = k + 2 * half;
        v2f a = *(const v2f*)(arow + kk);
        #pragma unroll
        for (int t = 0; t < 4; ++t) {
            v2f b = *(const v2f*)(WT + (size_t)(cn0 + t * 16 + lr) * CW + kk);
            acc[t] = wmma4(a, b, acc[t]);
        }
    }
    #pragma unroll
    for (int t = 0; t < 4; ++t) {
        const int cn = cn0 + t * 16;
        #pragma unroll
        for (int v = 0; v < 8; ++v)
            Out[(size_t)(rm + v + 8 * half) * CW + cn + lr] = acc[t][v];
    }
}

// ---------------------------------------------------------------------------
// CSR build (dst-ordered), reused across all 3 conv layers
// ---------------------------------------------------------------------------
__global__ void gnn_csr_zero_kernel(int* __restrict__ cnt) {
    int g = blockIdx.x * blockDim.x + threadIdx.x;
    if (g < NGRAPHS) cnt[g] = 0;
}

__global__ void gnn_csr_count_kernel(const int* __restrict__ dst,
                                     int* __restrict__ cnt) {
    int e = blockIdx.x * blockDim.x + threadIdx.x;
    if (e < NEDGES) atomicAdd(cnt + dst[e], 1);
}

__global__ void gnn_dis_kernel(const int* __restrict__ cnt,
                               float* __restrict__ dis) {
    int g = blockIdx.x * blockDim.x + threadIdx.x;
    if (g < NGRAPHS) dis[g] = rsqrtf((float)cnt[g] + 1.0f);  // +1 self loop
}

// Single-workgroup chunked exclusive scan: offs = exclusive_scan(cnt), cur = offs
__global__ __launch_bounds__(1024) void gnn_scan_kernel(
    const int* __restrict__ cnt, int* __restrict__ offs, int* __restrict__ cur) {
    __shared__ int sbuf[1024];
    __shared__ int carry;
    if (threadIdx.x == 0) carry = 0;
    __syncthreads();
    for (int base = 0; base < NGRAPHS; base += 1024) {
        const int i = base + (int)threadIdx.x;
        const int v = (i < NGRAPHS) ? cnt[i] : 0;
        sbuf[threadIdx.x] = v;
        __syncthreads();
        for (int off = 1; off < 1024; off <<= 1) {
            int t = (threadIdx.x >= (unsigned)off) ? sbuf[threadIdx.x - off] : 0;
            __syncthreads();
            sbuf[threadIdx.x] += t;
            __syncthreads();
        }
        const int incl = sbuf[threadIdx.x];
        const int c = carry;
        if (i < NGRAPHS) {
            const int o = c + incl - v;
            offs[i] = o;
            cur[i]  = o;
        }
        __syncthreads();
        if (threadIdx.x == 1023) carry = c + incl;
        __syncthreads();
    }
}

__global__ void gnn_csr_fill_kernel(const int* __restrict__ src,
                                    const int* __restrict__ dst,
                                    const float* __restrict__ dis,
                                    int* __restrict__ cur,
                                    int* __restrict__ csr_src,
                                    float* __restrict__ csr_nrm) {
    int e = blockIdx.x * blockDim.x + threadIdx.x;
    if (e >= NEDGES) return;
    const int s = src[e], d = dst[e];
    const int p = atomicAdd(cur + d, 1);
    csr_src[p] = s;
    csr_nrm[p] = dis[s] * dis[d];
}

// ---------------------------------------------------------------------------
// Gather: one wave per destination graph; 8 ch/lane in registers.
// Folds bias + self-loop + ELU.  No atomics; hw (51.2 MB) lives in L2.
// ---------------------------------------------------------------------------
__global__ __launch_bounds__(256) void gnn_gather_kernel(
    const float* __restrict__ hw, const int* __restrict__ offs,
    const int* __restrict__ cnt, const int* __restrict__ csr_src,
    const float* __restrict__ csr_nrm, const float* __restrict__ dis,
    const float* __restrict__ bias, float* __restrict__ hout) {
    const int g = blockIdx.x * 8 + (threadIdx.x >> 5);   // 6250 blocks exact
    const int lane = threadIdx.x & 31;
    const int c0 = lane * 8;

    const float dg   = dis[g];
    const float self = dg * dg;
    const float* hg  = hw + (size_t)g * CW + c0;
    v4f a0 = *(const v4f*)(bias + c0)     + self * *(const v4f*)(hg);
    v4f a1 = *(const v4f*)(bias + c0 + 4) + self * *(const v4f*)(hg + 4);

    const int s = offs[g];
    const int e = s + cnt[g];
    for (int i = s; i < e; ++i) {
        const int   sg = csr_src[i];
        const float nm = csr_nrm[i];
        const float* hs = hw + (size_t)sg * CW + c0;
        a0 += nm * *(const v4f*)(hs);
        a1 += nm * *(const v4f*)(hs + 4);
    }

    v4f r0, r1;
    #pragma unroll
    for (int j = 0; j < 4; ++j) { r0[j] = elu1(a0[j]); r1[j] = elu1(a1[j]); }
    float* o = hout + (size_t)g * CW + c0;
    *(v4f*)(o)     = r0;
    *(v4f*)(o + 4) = r1;
}

// ---------------------------------------------------------------------------
extern "C" void kernel_launch(void* const* d_in, const int* in_sizes, int n_in,
                              void* d_out, int out_size, void* d_ws,
                              size_t ws_size, hipStream_t stream) {
    const float* x  = (const float*)d_in[0];
    const int*   ei = (const int*)d_in[1];           // edge_index [2, 800000]
    const float* w2 = (const float*)d_in[2];
    const float* b2 = (const float*)d_in[3];
    const float* w1 = (const float*)d_in[4];
    const float* b1 = (const float*)d_in[5];
    const float* cw[3] = {(const float*)d_in[6], (const float*)d_in[8],
                          (const float*)d_in[10]};
    const float* cb[3] = {(const float*)d_in[7], (const float*)d_in[9],
                          (const float*)d_in[11]};
    float* out = (float*)d_out;

    float* ws   = (float*)d_ws;
    float* W0   = ws;                                   // 12.8M floats
    float* W1   = W0 + (size_t)NGRAPHS * CW;            // 12.8M floats
    float* dis  = W1 + (size_t)NGRAPHS * CW;            // 50k
    float* csr_nrm = dis + NGRAPHS;                     // 800k
    float* w2T  = csr_nrm + NEDGES;                     // 4096
    float* w1T  = w2T + INFEAT * FFN;                   // 2048
    float* cwT0 = w1T + FFN * HIDDEN;                   // 3 x 65536
    float* cwT[3] = {cwT0, cwT0 + CW * CW, cwT0 + 2 * CW * CW};
    int* cnt     = (int*)(cwT0 + 3 * CW * CW);          // 50k
    int* offs    = cnt + NGRAPHS;                       // 50k
    int* cur     = offs + NGRAPHS;                      // 50k
    int* csr_src = cur + NGRAPHS;                       // 800k

    const int* srcI = ei;
    const int* dstI = ei + NEDGES;

    // Weight transposes (tiny)
    gnn_transpose_kernel<<<(INFEAT * FFN + 255) / 256, 256, 0, stream>>>(w2, w2T, INFEAT, FFN);
    gnn_transpose_kernel<<<(FFN * HIDDEN + 255) / 256, 256, 0, stream>>>(w1, w1T, FFN, HIDDEN);
    for (int l = 0; l < 3; ++l)
        gnn_transpose_kernel<<<(CW * CW + 255) / 256, 256, 0, stream>>>(cw[l], cwT[l], CW, CW);

    // MLP front-end -> h0 in W0
    gnn_mlp_kernel<<<NGRAPHS / 4, 128, 0, stream>>>(x, w2T, b2, w1T, b1, W0);

    // CSR build (dst-ordered), degrees
    gnn_csr_zero_kernel<<<(NGRAPHS + 255) / 256, 256, 0, stream>>>(cnt);
    gnn_csr_count_kernel<<<(NEDGES + 255) / 256, 256, 0, stream>>>(dstI, cnt);
    gnn_dis_kernel<<<(NGRAPHS + 255) / 256, 256, 0, stream>>>(cnt, dis);
    gnn_scan_kernel<<<1, 1024, 0, stream>>>(cnt, offs, cur);
    gnn_csr_fill_kernel<<<(NEDGES + 255) / 256, 256, 0, stream>>>(
        srcI, dstI, dis, cur, csr_src, csr_nrm);

    // Layer 1: h=W0 -> hw=W1 -> gather(elu) -> out
    gnn_gemm256_kernel<<<NGRAPHS / 16, 128, 0, stream>>>(W0, cwT[0], W1);
    gnn_gather_kernel<<<NGRAPHS / 8, 256, 0, stream>>>(W1, offs, cnt, csr_src,
                                                       csr_nrm, dis, cb[0], out);
    // Layer 2: h=out -> hw=W1 -> gather -> W0
    gnn_gemm256_kernel<<<NGRAPHS / 16, 128, 0, stream>>>(out, cwT[1], W1);
    gnn_gather_kernel<<<NGRAPHS / 8, 256, 0, stream>>>(W1, offs, cnt, csr_src,
                                                       csr_nrm, dis, cb[1], W0);
    // Layer 3: h=W0 -> hw=W1 -> gather -> out (final)
    gnn_gemm256_kernel<<<NGRAPHS / 16, 128, 0, stream>>>(W0, cwT[2], W1);
    gnn_gather_kernel<<<NGRAPHS / 8, 256, 0, stream>>>(W1, offs, cnt, csr_src,
                                                       csr_nrm, dis, cb[2], out);
}